// XTransEncoderAct_19576460935822
// MI455X (gfx1250) — compile-verified
//
#include <hip/hip_runtime.h>
#include <math.h>

// ---------------------------------------------------------------------------
// MI455X (gfx1250) fused transformer-encoder layer, fp32 via
// V_WMMA_F32_16X16X4_F32 (exact fp32 semantics; reference is fp32).
// B=4, N=1024, D=512, H=16, DH=32, DFF=2048.
// GEMM: 32x32 register blocking per wave -> 1:1 WMMA:load ratio.
// ---------------------------------------------------------------------------

typedef __attribute__((ext_vector_type(2))) float v2f;
typedef __attribute__((ext_vector_type(8))) float v8f;

#define BB   4
#define NN   1024
#define DD   512
#define HH   16
#define DHD  32
#define ATT_FACTOR 0.08838834764831845f   // 0.5 / sqrt(32)

#define KC     128                         // GEMM K-chunk staged in LDS
#define LDSTR  132                         // padded row stride (conflict-free)

static __device__ __forceinline__ v8f wmma_f32(v2f a, v2f b, v8f c) {
    // D(16x16,f32) = A(16x4,f32) x B(4x16,f32) + C
    return __builtin_amdgcn_wmma_f32_16x16x4_f32(
        /*neg_a=*/false, a, /*neg_b=*/false, b,
        /*c_mod=*/(short)0, c, /*reuse_a=*/false, /*reuse_b=*/false);
}

// ---------------------------------------------------------------------------
// Generic WMMA GEMM: out = X[M,K] @ W[Nn,K]^T (+bias) with epilogue modes.
//   mode 0: out[row*Nn+col] = acc + bias
//   mode 1: SiLU(acc + bias)
//   mode 2: resid[row*Nn+col] + acc + bias        (residual add)
//   mode 3: scatter Hv into V_attn[...,0:32]      (row over B*N)
//   mode 4: scatter Vv into V_attn[...,32:128]    (row over B*N*3)
// One wave -> 32x32 output (2x2 fragments, 4 accumulators); 8 waves (2x4)
// -> 64x128 block tile. A tile staged via padded LDS (64 x KC, stride 132).
// Inner loop: 2 LDS b64 + 2 global b64 loads feed 4 WMMAs.
// Requires: M % 64 == 0, Nn % 128 == 0, K % KC == 0.
// ---------------------------------------------------------------------------
__global__ void __launch_bounds__(256)
gemm_wmma_kernel(const float* __restrict__ X, const float* __restrict__ W,
                 const float* __restrict__ bias, const float* __restrict__ resid,
                 float* __restrict__ out, int M, int Nn, int K, int mode)
{
    extern __shared__ float ldsA[];            // 64 * LDSTR floats (33 KB)
    const int tid  = threadIdx.x;
    const int lane = tid & 31;
    const int w    = tid >> 5;
    const int l15  = lane & 15;
    const int koff = (lane >> 4) * 2;
    const int mgrp = w >> 2;                   // 0..1 : wave M sub-block
    const int ngrp = w & 3;                    // 0..3 : wave N sub-block

    const int row0 = blockIdx.x * 64;
    const int col0 = blockIdx.y * 128;
    const int c0   = col0 + ngrp * 32 + l15;   // B columns for this lane
    const int c1   = c0 + 16;

    v8f acc[2][2] = {{{}, {}}, {{}, {}}};

    const float* ap0base = ldsA + (mgrp * 32 + l15) * LDSTR + koff;
    const float* ap1base = ap0base + 16 * LDSTR;

    for (int kc = 0; kc < K; kc += KC) {
        __syncthreads();
        for (int i = tid; i < 64 * KC; i += 256) {
            int r  = i >> 7;                   // KC == 128
            int kk = i & (KC - 1);
            ldsA[r * LDSTR + kk] = X[(long)(row0 + r) * K + kc + kk];
        }
        __syncthreads();

        const float* wp0 = W + (long)c0 * K + kc + koff;
        const float* wp1 = W + (long)c1 * K + kc + koff;
        if (kc + KC < K) {
            __builtin_prefetch(wp0 + KC, 0, 1);   // global_prefetch_b8
            __builtin_prefetch(wp1 + KC, 0, 1);
        }

        #pragma unroll 4
        for (int k = 0; k < KC; k += 4) {
            v2f a0 = *(const v2f*)(ap0base + k);   // A: lane=M row, halfwave=K
            v2f a1 = *(const v2f*)(ap1base + k);
            v2f b0 = *(const v2f*)(wp0 + k);       // B: lane=N col, halfwave=K
            v2f b1 = *(const v2f*)(wp1 + k);
            acc[0][0] = wmma_f32(a0, b0, acc[0][0]);
            acc[0][1] = wmma_f32(a0, b1, acc[0][1]);
            acc[1][0] = wmma_f32(a1, b0, acc[1][0]);
            acc[1][1] = wmma_f32(a1, b1, acc[1][1]);
        }
    }

    #pragma unroll
    for (int mi = 0; mi < 2; ++mi) {
        #pragma unroll
        for (int ni = 0; ni < 2; ++ni) {
            const int colw = col0 + ngrp * 32 + ni * 16 + l15;
            const float bval = bias ? bias[colw] : 0.0f;
            #pragma unroll
            for (int r = 0; r < 8; ++r) {
                const int row = row0 + mgrp * 32 + mi * 16 + r + 8 * (lane >> 4);
                float v = acc[mi][ni][r];
                switch (mode) {
                case 0:
                    out[(long)row * Nn + colw] = v + bval;
                    break;
                case 1: {
                    float t = v + bval;
                    out[(long)row * Nn + colw] = t / (1.0f + expf(-t)); // SiLU
                }   break;
                case 2:
                    out[(long)row * Nn + colw] =
                        resid[(long)row * Nn + colw] + v + bval;
                    break;
                case 3: {   // Hv -> V_attn[b,n,h,0:32]
                    int h = colw >> 5, d = colw & 31;
                    out[((long)row * HH + h) * 128 + d] = v + bval;
                }   break;
                case 4: {   // Vv -> V_attn[b,n,h,32+c*32+d]  (row over B*N*3)
                    int bn = row / 3, c = row - bn * 3;
                    int h = colw >> 5, d = colw & 31;
                    out[((long)bn * HH + h) * 128 + 32 + c * 32 + d] = v;
                }   break;
                }
            }
        }
    }
}

// ---------------------------------------------------------------------------
// LayerNorm over last dim (512). One block per row.
// ---------------------------------------------------------------------------
__global__ void __launch_bounds__(256)
layernorm_kernel(const float* __restrict__ X, const float* __restrict__ g,
                 const float* __restrict__ bta, float* __restrict__ out, int ldo)
{
    const int row = blockIdx.x;
    const int tid = threadIdx.x;
    const float* x = X + (long)row * DD;
    float x0 = x[tid], x1 = x[tid + 256];
    float s = x0 + x1, ss = x0 * x0 + x1 * x1;
    #pragma unroll
    for (int off = 16; off >= 1; off >>= 1) {
        s  += __shfl_xor(s, off, 32);
        ss += __shfl_xor(ss, off, 32);
    }
    __shared__ float rs[8], rss[8];
    if ((tid & 31) == 0) { rs[tid >> 5] = s; rss[tid >> 5] = ss; }
    __syncthreads();
    s = 0.f; ss = 0.f;
    #pragma unroll
    for (int i = 0; i < 8; ++i) { s += rs[i]; ss += rss[i]; }
    const float mean = s * (1.0f / DD);
    const float var  = ss * (1.0f / DD) - mean * mean;
    const float inv  = rsqrtf(var + 1e-5f);
    out[(long)row * ldo + tid]       = (x0 - mean) * inv * g[tid]       + bta[tid];
    out[(long)row * ldo + tid + 256] = (x1 - mean) * inv * g[tid + 256] + bta[tid + 256];
}

// ---------------------------------------------------------------------------
// Attention: one block per (b, h, 16-query tile).
// Phase 1: logits (16 x 1024) via WMMA over K=128; two key tiles per pass
//          share one Q A-fragment load. Bias/mask fused -> LDS.
// Phase 2: row softmax (width-16 shuffle reductions, no extra LDS).
// Phase 3: P @ V_attn via WMMA over K=1024 -> scatter to H_r / V_r layouts.
// P is XOR-swizzled (n ^ (row<<2)) for bank-conflict-free phase-3 A reads.
// ---------------------------------------------------------------------------
__global__ void __launch_bounds__(256)
attn_kernel(const float* __restrict__ Hq, const float* __restrict__ Hk,
            const float* __restrict__ Vattn, const float* __restrict__ rbf,
            const float* __restrict__ abp, const unsigned char* __restrict__ mask,
            float* __restrict__ Hr, float* __restrict__ Vr)
{
    __shared__ float P[16 * 1024];            // 64 KB probability tile
    const int b  = blockIdx.z, h = blockIdx.y, qt = blockIdx.x;
    const int tid = threadIdx.x, lane = tid & 31, w = tid >> 5;
    const int l15 = lane & 15, koff = (lane >> 4) * 2;

    const float apv = abp[h];
    const float* qp = Hq + (((long)b * NN + qt * 16 + l15) * HH + h) * 128 + koff;

    // ---- Phase 1: logits; wave w covers key tiles [w*8, w*8+8) in pairs ----
    for (int p = 0; p < 4; ++p) {
        const int nt0 = w * 8 + p * 2;
        v8f acc0 = {}, acc1 = {};
        const float* kp0 = Hk + (((long)b * NN + nt0 * 16 + l15) * HH + h) * 128 + koff;
        const float* kp1 = kp0 + (long)16 * HH * 128;   // next 16 keys
        #pragma unroll 8
        for (int k = 0; k < 128; k += 4) {
            v2f a  = *(const v2f*)(qp + k);             // shared Q fragment
            v2f b0 = *(const v2f*)(kp0 + k);
            v2f b1 = *(const v2f*)(kp1 + k);
            acc0 = wmma_f32(a, b0, acc0);
            acc1 = wmma_f32(a, b1, acc1);
        }
        #pragma unroll
        for (int t = 0; t < 2; ++t) {
            const v8f accT = t ? acc1 : acc0;
            const int ncol = (nt0 + t) * 16 + l15;
            const bool mk  = mask[b * NN + ncol] != 0;
            #pragma unroll
            for (int r = 0; r < 8; ++r) {
                const int row  = r + 8 * (lane >> 4);
                const int qrow = qt * 16 + row;
                float lg = mk ? (accT[r] * ATT_FACTOR
                                 + rbf[((long)b * NN + qrow) * NN + ncol] + apv)
                              : -INFINITY;
                P[row * 1024 + (ncol ^ (row << 2))] = lg;
            }
        }
    }
    __syncthreads();

    // ---- Phase 2: softmax over each of the 16 rows ----
    {
        const int r = tid >> 4, g = tid & 15;
        float* prow = P + r * 1024;
        const int sw = r << 2;
        float m = -INFINITY;
        for (int j = g; j < 1024; j += 16) m = fmaxf(m, prow[j ^ sw]);
        #pragma unroll
        for (int off = 8; off >= 1; off >>= 1) m = fmaxf(m, __shfl_xor(m, off, 16));
        float ssum = 0.f;
        for (int j = g; j < 1024; j += 16) {
            float e = expf(prow[j ^ sw] - m);
            prow[j ^ sw] = e;
            ssum += e;
        }
        #pragma unroll
        for (int off = 8; off >= 1; off >>= 1) ssum += __shfl_xor(ssum, off, 16);
        const float invs = 1.0f / ssum;
        for (int j = g; j < 1024; j += 16) prow[j ^ sw] *= invs;
    }
    __syncthreads();

    // ---- Phase 3: out(16 x 128) = P(16x1024) @ V_attn(1024x128) ----
    v8f acc = {};
    const float* prowA = P + l15 * 1024;      // A row = query row = l15
    const int swA = l15 << 2;
    const int d = w * 16 + l15;               // output feature column (0..127)
    const float* vb = Vattn + ((long)b * NN * HH + h) * 128 + d;
    for (int kk = 0; kk < 1024; kk += 4) {
        const int k0 = kk + koff;
        v2f a = *(const v2f*)(prowA + (k0 ^ swA));  // swizzle keeps pairs adjacent
        v2f bb;
        bb.x = vb[(long)k0 * (HH * 128)];
        bb.y = vb[(long)(k0 + 1) * (HH * 128)];
        acc = wmma_f32(a, bb, acc);
    }

    #pragma unroll
    for (int r = 0; r < 8; ++r) {
        const int qrow = qt * 16 + r + 8 * (lane >> 4);
        const long n = (long)b * NN + qrow;
        const float v = acc[r];
        if (d < DHD) {
            Hr[n * DD + h * DHD + d] = v;                          // res[...,:32]
        } else {
            const int c = (d - DHD) >> 5, dd = (d - DHD) & 31;     // res[...,32:]
            Vr[(n * 3 + c) * DD + h * DHD + dd] = v;
        }
    }
}

// ---------------------------------------------------------------------------
// v1_norm: scaler[:,512+d] = || V1[:, :, d] ||_2 over the 3-axis.
// ---------------------------------------------------------------------------
__global__ void __launch_bounds__(256)
v1norm_kernel(const float* __restrict__ Vp, float* __restrict__ scaler)
{
    const int idx = blockIdx.x * 256 + threadIdx.x;      // over 4096*512
    const int row = idx >> 9, d = idx & 511;
    const long base = (long)row * 3 * 1024 + d;
    float a = Vp[base], b = Vp[base + 1024], c = Vp[base + 2048];
    scaler[(long)row * 1024 + DD + d] = sqrtf(a * a + b * b + c * c);
}

// ---------------------------------------------------------------------------
// Final combine: H = H1 + so[:, :512] ;  V = V1 + so[:, 512:][...,None,:] * V2
// ---------------------------------------------------------------------------
__global__ void __launch_bounds__(256)
final_kernel(const float* __restrict__ H1, const float* __restrict__ V1,
             const float* __restrict__ so, const float* __restrict__ Vp,
             float* __restrict__ outH, float* __restrict__ outV)
{
    const int idx = blockIdx.x * 256 + threadIdx.x;
    const int HTOT = BB * NN * DD;                       // 2,097,152
    if (idx < HTOT) {
        const int row = idx >> 9, d = idx & 511;
        outH[idx] = H1[idx] + so[(long)row * 1024 + d];
    } else {
        const int j = idx - HTOT;                        // over B*N*3*D
        const int d = j & 511;
        const int c = (j >> 9) % 3;
        const int row = j / (3 * DD);
        const float vu = so[(long)row * 1024 + DD + d];
        const float v2 = Vp[((long)row * 3 + c) * 1024 + DD + d];
        outV[j] = V1[j] + vu * v2;
    }
}

// ---------------------------------------------------------------------------
extern "C" void kernel_launch(void* const* d_in, const int* in_sizes, int n_in,
                              void* d_out, int out_size, void* d_ws, size_t ws_size,
                              hipStream_t stream)
{
    const float* H    = (const float*)d_in[0];
    const float* V    = (const float*)d_in[1];
    const unsigned char* mask = (const unsigned char*)d_in[2];
    const float* rbf  = (const float*)d_in[3];
    const float* abp  = (const float*)d_in[4];
    const float* Wq   = (const float*)d_in[5];
    const float* bq   = (const float*)d_in[6];
    const float* Wk   = (const float*)d_in[7];
    const float* bk   = (const float*)d_in[8];
    const float* Wv   = (const float*)d_in[9];
    const float* bv   = (const float*)d_in[10];
    const float* Wvv  = (const float*)d_in[11];
    const float* Wo   = (const float*)d_in[12];
    const float* bo   = (const float*)d_in[13];
    const float* Wvo  = (const float*)d_in[14];
    const float* ln1g = (const float*)d_in[15];
    const float* ln1b = (const float*)d_in[16];
    const float* ln2g = (const float*)d_in[17];
    const float* ln2b = (const float*)d_in[18];
    const float* Wlv  = (const float*)d_in[19];
    const float* W1   = (const float*)d_in[20];
    const float* b1   = (const float*)d_in[21];
    const float* W2   = (const float*)d_in[22];
    const float* b2   = (const float*)d_in[23];

    float* ws = (float*)d_ws;
    const long M1 = 1024 * 1024;
    // Phase-A buffers                                   (peak ws: 42M floats = 168 MB)
    float* Hn    = ws + 0 * M1;    // (4096,512)
    float* Hqb   = ws + 2 * M1;    // (4096,2048)
    float* Hkb   = ws + 10 * M1;   // (4096,2048)
    float* Vatt  = ws + 18 * M1;   // (B,N,H,128)
    float* Hrb   = ws + 26 * M1;   // (4096,512)
    float* Vrb   = ws + 28 * M1;   // (12288,512)
    float* H1b   = ws + 34 * M1;   // (4096,512)   live to end
    float* V1b   = ws + 36 * M1;   // (12288,512)  live to end
    // Phase-B buffers (alias region freed after attention/projections)
    float* scal  = ws + 0 * M1;    // (4096,1024)
    float* Vpb   = ws + 4 * M1;    // (12288,1024)
    float* h1b   = ws + 16 * M1;   // (4096,2048)
    float* sob   = ws + 24 * M1;   // (4096,1024)

    const size_t SH = 64 * LDSTR * sizeof(float);   // 33 KB padded A-tile

    // 1) LN1
    layernorm_kernel<<<4096, 256, 0, stream>>>(H, ln1g, ln1b, Hn, DD);
    // 2-3) Q, K projections (B*N x 2048)
    gemm_wmma_kernel<<<dim3(64, 16), 256, SH, stream>>>(Hn, Wq, bq, nullptr, Hqb, 4096, 2048, 512, 0);
    gemm_wmma_kernel<<<dim3(64, 16), 256, SH, stream>>>(Hn, Wk, bk, nullptr, Hkb, 4096, 2048, 512, 0);
    // 4) Hv -> V_attn[...,0:32]
    gemm_wmma_kernel<<<dim3(64, 4), 256, SH, stream>>>(Hn, Wv, bv, nullptr, Vatt, 4096, 512, 512, 3);
    // 5) Vv -> V_attn[...,32:128]
    gemm_wmma_kernel<<<dim3(192, 4), 256, SH, stream>>>(V, Wvv, nullptr, nullptr, Vatt, 12288, 512, 512, 4);
    // 6) attention (logits + softmax + P@V), writes H_r / V_r layouts
    attn_kernel<<<dim3(64, HH, BB), 256, 0, stream>>>(Hqb, Hkb, Vatt, rbf, abp, mask, Hrb, Vrb);
    // 7-8) output projections with fused residuals
    gemm_wmma_kernel<<<dim3(64, 4), 256, SH, stream>>>(Hrb, Wo, bo, H, H1b, 4096, 512, 512, 2);
    gemm_wmma_kernel<<<dim3(192, 4), 256, SH, stream>>>(Vrb, Wvo, nullptr, V, V1b, 12288, 512, 512, 2);
    // 9) LN2 -> scaler[:, :512]
    layernorm_kernel<<<4096, 256, 0, stream>>>(H1b, ln2g, ln2b, scal, 1024);
    // 10) V_proj = V1b @ Wlv^T  (12288 x 1024)
    gemm_wmma_kernel<<<dim3(192, 8), 256, SH, stream>>>(V1b, Wlv, nullptr, nullptr, Vpb, 12288, 1024, 512, 0);
    // 11) v1 norm -> scaler[:, 512:]
    v1norm_kernel<<<8192, 256, 0, stream>>>(Vpb, scal);
    // 12) h1 = SiLU(scaler @ W1^T + b1)
    gemm_wmma_kernel<<<dim3(64, 16), 256, SH, stream>>>(scal, W1, b1, nullptr, h1b, 4096, 2048, 1024, 1);
    // 13) scaler_out = h1 @ W2^T + b2
    gemm_wmma_kernel<<<dim3(64, 8), 256, SH, stream>>>(h1b, W2, b2, nullptr, sob, 4096, 1024, 2048, 0);
    // 14) final combine -> d_out = [H | V]
    float* outF = (float*)d_out;
    final_kernel<<<32768, 256, 0, stream>>>(H1b, V1b, sob, Vpb, outF, outF + (long)BB * NN * DD);
}